// LightGCN_50964081935596
// MI455X (gfx1250) — compile-verified
//
#include <hip/hip_runtime.h>

typedef float v2f __attribute__((ext_vector_type(2)));
typedef float v8f __attribute__((ext_vector_type(8)));

#define EMBED 64

// ---------------------------------------------------------------------------
// Guaranteed-hardware fp32 atomic add: GLOBAL_ATOMIC_ADD_F32, non-returning
// (STOREcnt-tracked, no pre-op return). Avoids any CAS-loop expansion the
// generic __hip_atomic_fetch_add path might take for fp32.
// ---------------------------------------------------------------------------
__device__ __forceinline__ void atomic_add_f32_hw(float* p, float v) {
  unsigned long long a = (unsigned long long)p;
  asm volatile("global_atomic_add_f32 %0, %1, off"
               :
               : "v"(a), "v"(v)
               : "memory");
}

// ---------------------------------------------------------------------------
// Phase 1: deg[i] = 1 (self loop) + count of edges with row==i
// ---------------------------------------------------------------------------
__global__ void k_init_deg(float* __restrict__ deg, int N) {
  int i = blockIdx.x * blockDim.x + threadIdx.x;
  if (i < N) deg[i] = 1.0f;  // self-loop contribution
}

__global__ void k_count(const int* __restrict__ row, long long E,
                        float* __restrict__ deg) {
  long long i = (long long)blockIdx.x * blockDim.x + threadIdx.x;
  long long base = i * 4;
  if (base + 3 < E) {
    int4 r = *(const int4*)(row + base);
    atomic_add_f32_hw(&deg[r.x], 1.0f);
    atomic_add_f32_hw(&deg[r.y], 1.0f);
    atomic_add_f32_hw(&deg[r.z], 1.0f);
    atomic_add_f32_hw(&deg[r.w], 1.0f);
  } else if (base < E) {
    for (long long e = base; e < E; ++e)
      atomic_add_f32_hw(&deg[row[e]], 1.0f);
  }
}

// ---------------------------------------------------------------------------
// Phase 2: per-node coefficients.
//   inv_deg = 1/(deg+1e-8); deg2 = deg*inv_deg; s = rsqrt(deg2+1e-8)
//   edge weight(row,col) = a[row]*b[col], a = s*inv_deg, b = s
// ---------------------------------------------------------------------------
__global__ void k_coef(const float* __restrict__ deg,
                       float* __restrict__ ca, float* __restrict__ cb, int N) {
  int i = blockIdx.x * blockDim.x + threadIdx.x;
  if (i >= N) return;
  float d = deg[i];
  float inv_deg = 1.0f / (d + 1e-8f);
  float deg2 = d * inv_deg;
  float s = (deg2 > 0.0f) ? rsqrtf(deg2 + 1e-8f) : 0.0f;
  ca[i] = s * inv_deg;
  cb[i] = s;
}

// ---------------------------------------------------------------------------
// Phase 3: h = diag(a.*b) * x  (exact self-loop init) via V_WMMA_F32_16X16X4_F32.
// One wave owns one 16(nodes) x 16(features) tile of h. diag(c)@X is computed
// exactly as sum_{j=0..3} A_j(16x4) @ X_j(4x16), where A_j holds columns
// 4j..4j+3 of diag(c) (fp32, bit-exact — each output element is one multiply).
//
// VGPR layouts (ISA 7.12.2, 32-bit, wave32):
//   A 16x4 : lane[0..15] M=lane, v0:K=0 v1:K=1 ; lane[16..31] M=lane-16, v0:K=2 v1:K=3
//   B 4x16 : mirrored (N=lane&15, K split {0,1}/{2,3} across lane halves)
//   C 16x16: vr -> M=r (lanes 0-15, N=lane) / M=r+8 (lanes 16-31, N=lane-16)
// ---------------------------------------------------------------------------
__global__ void k_init_h(const float* __restrict__ x,
                         const float* __restrict__ ca,
                         const float* __restrict__ cb,
                         float* __restrict__ h, int N) {
  int lane = threadIdx.x & 31;
  int wave = threadIdx.x >> 5;
  int task = blockIdx.x * 8 + wave;
  int ntiles = (N + 15) >> 4;
  if (task >= ntiles * 4) return;
  int tile = task >> 2;
  int feat = (task & 3) << 4;   // which 16-wide feature slab of the 64
  int base = tile << 4;         // first node of the tile
  int m  = lane & 15;
  int hi = lane >> 4;

  if (base + 16 <= N) {
    int node = base + m;
    float cm = ca[node] * cb[node];   // self-loop weight for row m
    v8f acc = {};
#pragma unroll
    for (int j = 0; j < 4; ++j) {
      int r0 = 4 * j + (hi ? 2 : 0);  // K indices handled by this lane half
      int r1 = r0 + 1;
      v2f a, b;
      a.x = (m == r0) ? cm : 0.0f;    // column r0 of diag(c)
      a.y = (m == r1) ? cm : 0.0f;    // column r1 of diag(c)
      b.x = x[(long long)(base + r0) * EMBED + feat + m];  // row r0 of X tile
      b.y = x[(long long)(base + r1) * EMBED + feat + m];  // row r1 of X tile
      acc = __builtin_amdgcn_wmma_f32_16x16x4_f32(
          /*neg_a=*/false, a, /*neg_b=*/false, b,
          /*c_mod=*/(short)0, acc, /*reuse_a=*/false, /*reuse_b=*/false);
    }
#pragma unroll
    for (int r = 0; r < 8; ++r) {
      int rowm = hi ? (r + 8) : r;
      h[(long long)(base + rowm) * EMBED + feat + m] = acc[r];
    }
  } else {
    // ragged tail tile: scalar path (WMMA needs EXEC all-ones)
    int node = base + m;
    if (hi == 0 && node < N) {
      float cm = ca[node] * cb[node];
#pragma unroll
      for (int f = 0; f < 16; ++f)
        h[(long long)node * EMBED + feat + f] =
            cm * x[(long long)node * EMBED + feat + f];
    }
  }
}

// ---------------------------------------------------------------------------
// Phase 4: edge scatter. One wave per edge: wave-uniform row/col/weight go
// through the scalar path (readfirstlane), each lane moves 2 floats of the
// 256B x[col] row (coalesced b64 load, L2-resident) and issues 2 hardware
// GLOBAL_ATOMIC_ADD_F32 ops.
// ---------------------------------------------------------------------------
__global__ void k_scatter(const float* __restrict__ x,
                          const int* __restrict__ erow,
                          const int* __restrict__ ecol,
                          const float* __restrict__ ca,
                          const float* __restrict__ cb,
                          float* __restrict__ h, long long E) {
  int lane = threadIdx.x & 31;
  int wave = __builtin_amdgcn_readfirstlane((int)(threadIdx.x >> 5));
  long long e = (long long)blockIdx.x * 8 + wave;
  if (e >= E) return;

  // prefetch ahead on the sequential edge stream (gfx1250 global_prefetch_b8)
  if (lane == 0 && e + 2048 < E) {
    __builtin_prefetch(erow + e + 2048, 0, 1);
    __builtin_prefetch(ecol + e + 2048, 0, 1);
  }

  int r = erow[e];               // wave-uniform -> scalar path
  int c = ecol[e];
  float w = ca[r] * cb[c];       // normalized edge weight

  const v2f xv = *(const v2f*)(x + (long long)c * EMBED + lane * 2);
  float* hp = h + (long long)r * EMBED + lane * 2;
  atomic_add_f32_hw(hp,     w * xv.x);
  atomic_add_f32_hw(hp + 1, w * xv.y);
}

// ---------------------------------------------------------------------------
extern "C" void kernel_launch(void* const* d_in, const int* in_sizes, int n_in,
                              void* d_out, int out_size, void* d_ws, size_t ws_size,
                              hipStream_t stream) {
  const float* x    = (const float*)d_in[0];
  const int*   edge = (const int*)d_in[1];   // [2,E] flat: rows then cols
  int       N = in_sizes[0] / EMBED;
  long long E = (long long)in_sizes[1] / 2;

  float* h   = (float*)d_out;
  float* ca  = (float*)d_ws;        // N floats
  float* cb  = ca + N;              // N floats
  float* deg = cb + N;              // N floats  (total 3N*4 = 2.4 MB)

  const int T = 256;

  k_init_deg<<<(N + T - 1) / T, T, 0, stream>>>(deg, N);

  long long cthreads = (E + 3) / 4;
  if (cthreads > 0)
    k_count<<<(unsigned)((cthreads + T - 1) / T), T, 0, stream>>>(edge, E, deg);

  k_coef<<<(N + T - 1) / T, T, 0, stream>>>(deg, ca, cb, N);

  int tasks = ((N + 15) / 16) * 4;          // 16-node x 16-feat tiles
  k_init_h<<<(tasks + 7) / 8, T, 0, stream>>>(x, ca, cb, h, N);

  if (E > 0)
    k_scatter<<<(unsigned)((E + 7) / 8), T, 0, stream>>>(x, edge, edge + E,
                                                         ca, cb, h, E);
}